// FRC_1829656068367
// MI455X (gfx1250) — compile-verified
//
#include <hip/hip_runtime.h>
#include <math.h>

typedef float v2f __attribute__((ext_vector_type(2)));
typedef float v8f __attribute__((ext_vector_type(8)));

#define BN_EPS 1e-5f

// Problem dims (fixed by setup_inputs)
constexpr int NB  = 8;
constexpr int C   = 64;
constexpr int HIN = 64,  WIN = 64;
constexpr int H   = 128, W   = 128;
constexpr int PIN = HIN * WIN;   // 4096 pixels per batch (low res)
constexpr int P   = H * W;       // 16384 pixels per batch (high res)

// ---------------------------------------------------------------------------
// Kernel 0: res[b,h,w] = mean_c ref[b,c,h,w]
// ---------------------------------------------------------------------------
__global__ void res_mean_kernel(const float* __restrict__ ref,
                                float* __restrict__ res) {
    int idx = blockIdx.x * blockDim.x + threadIdx.x;   // over NB*P (exact grid)
    int b = idx >> 14;          // /P
    int p = idx & (P - 1);
    const float* rp = ref + (size_t)b * C * P + p;
    float s = 0.f;
#pragma unroll 8
    for (int c = 0; c < C; ++c) s += rp[c * P];
    res[idx] = s * (1.0f / 64.0f);
}

// ---------------------------------------------------------------------------
// Kernel 1: y = relu(BN(conv1x1(x)))  via V_WMMA_F32_16X16X4_F32
// One wave computes a 16(out-ch) x 16(pixel) tile; K=64 in 16 steps of 4.
// BN scale is folded into the A (weight) rows; bias applied post-accumulate.
// ---------------------------------------------------------------------------
__global__ void conv1_wmma_kernel(const float* __restrict__ x,
                                  const float* __restrict__ w1,
                                  const float* __restrict__ b1,
                                  const float* __restrict__ g1,
                                  const float* __restrict__ beta1,
                                  const float* __restrict__ m1,
                                  const float* __restrict__ var1,
                                  float* __restrict__ y) {
    const int lane = threadIdx.x & 31;
    const int wid  = (blockIdx.x * blockDim.x + threadIdx.x) >> 5;
    // tiles: b in [0,8), mt in [0,4), pt in [0,256)
    const int b  = wid >> 10;
    const int mt = (wid >> 8) & 3;
    const int pt = wid & 255;

    const int half = lane >> 4;       // 0: lanes 0-15, 1: lanes 16-31
    const int lm   = lane & 15;
    const int kSel = half * 2;        // A/B K sub-select per ISA layout

    // A row for this lane (same for both halves): out channel mt*16+lm
    const int   oA = mt * 16 + lm;
    const float sA = g1[oA] * rsqrtf(var1[oA] + BN_EPS);
    const float* wrow = w1 + oA * 64;

    // B column for this lane: pixel pt*16+lm, stride PIN between K rows
    const float* xcol = x + (size_t)b * C * PIN + pt * 16 + lm;

    v8f acc = {};
#pragma unroll
    for (int kk = 0; kk < 64; kk += 4) {
        v2f a, bm;
        a.x  = wrow[kk + kSel]     * sA;
        a.y  = wrow[kk + kSel + 1] * sA;
        bm.x = xcol[(kk + kSel)     * PIN];
        bm.y = xcol[(kk + kSel + 1) * PIN];
        acc = __builtin_amdgcn_wmma_f32_16x16x4_f32(
                  false, a, false, bm, (short)0, acc, false, false);
    }

    // bias + ReLU + store. C/D layout: VGPR v -> row v (lanes 0-15), v+8 (16-31)
    float* yb = y + (size_t)b * C * PIN + pt * 16 + lm;
#pragma unroll
    for (int v = 0; v < 8; ++v) {
        const int   o    = mt * 16 + v + half * 8;
        const float s    = g1[o] * rsqrtf(var1[o] + BN_EPS);
        const float bias = b1[o] * s + beta1[o] - m1[o] * s;
        yb[o * PIN] = fmaxf(acc[v] + bias, 0.f);
    }
}

// ---------------------------------------------------------------------------
// Kernel 2: fused conv2(WMMA)+BN+ReLU  +  mask  +  guided average  +  add
// One wave computes a 16(out-ch) x 16(pixel) tile of the final output.
// Because resize is exactly 2x nearest, the 3x3 window over y_up touches at
// most 2x2 distinct y elements -> collapse the 9 mask weights into 4.
// ---------------------------------------------------------------------------
__global__ void fused_out_kernel(const float* __restrict__ ref,
                                 const float* __restrict__ w2,
                                 const float* __restrict__ b2,
                                 const float* __restrict__ g2,
                                 const float* __restrict__ beta2,
                                 const float* __restrict__ m2,
                                 const float* __restrict__ var2,
                                 const float* __restrict__ res,
                                 const float* __restrict__ y,
                                 float* __restrict__ out) {
    const int lane = threadIdx.x & 31;
    const int wid  = (blockIdx.x * blockDim.x + threadIdx.x) >> 5;
    // tiles: b in [0,8), ct in [0,4), pt in [0,1024)
    const int b  = wid >> 12;
    const int ct = (wid >> 10) & 3;
    const int pt = wid & 1023;

    const int half = lane >> 4;
    const int lm   = lane & 15;
    const int kSel = half * 2;

    // ---- conv2 GEMM (WMMA) ----
    const int   oA = ct * 16 + lm;
    const float sA = g2[oA] * rsqrtf(var2[oA] + BN_EPS);
    const float* wrow = w2 + oA * 64;
    const float* rcol = ref + (size_t)b * C * P + pt * 16 + lm;

    v8f acc = {};
#pragma unroll
    for (int kk = 0; kk < 64; kk += 4) {
        v2f a, bm;
        a.x  = wrow[kk + kSel]     * sA;
        a.y  = wrow[kk + kSel + 1] * sA;
        bm.x = rcol[(kk + kSel)     * P];
        bm.y = rcol[(kk + kSel + 1) * P];
        acc = __builtin_amdgcn_wmma_f32_16x16x4_f32(
                  false, a, false, bm, (short)0, acc, false, false);
    }

    // ---- per-lane mask for this lane's pixel (column of the tile) ----
    const int p  = pt * 16 + lm;
    const int ph = p >> 7;        // /W
    const int pw = p & 127;
    const float* resb = res + b * P;

    float ur[9];
    float ssum = 0.f;
#pragma unroll
    for (int di = 0; di < 3; ++di)
#pragma unroll
        for (int dj = 0; dj < 3; ++dj) {
            const int hh = ph + di - 1, ww = pw + dj - 1;
            const bool ok = (hh >= 0) & (hh < H) & (ww >= 0) & (ww < W);
            const float v = ok ? resb[hh * W + ww] : 0.f;
            ur[di * 3 + dj] = v;
            ssum += v;
        }
    const float ua   = ssum * (1.f / 9.f);
    const float ui   = (ur[4] - ua > 0.f) ? 1.f : 0.f;

    // 2x nearest upsample: neighbor rows map to {r1-1, r1}, cols to {c1-1, c1}
    const int r1 = (ph + 1) >> 1, c1 = (pw + 1) >> 1;
    const int r0c = max(r1 - 1, 0), r1c = min(r1, HIN - 1);
    const int c0c = max(c1 - 1, 0), c1c = min(c1, WIN - 1);

    float wgt00 = 0.f, wgt01 = 0.f, wgt10 = 0.f, wgt11 = 0.f;
    float den = 1e-6f;
#pragma unroll
    for (int di = 0; di < 3; ++di)
#pragma unroll
        for (int dj = 0; dj < 3; ++dj) {
            const int k = di * 3 + dj;
            const float up = (ur[k] - ua > 0.f) ? 1.f : 0.f;
            const float mk = ui * up + (1.f - ui) * (1.f - up);
            den += mk;                                    // den counts all 9
            const int hh = ph + di - 1, ww = pw + dj - 1;
            if (hh >= 0 && hh < H && ww >= 0 && ww < W) { // num drops OOB (u==0)
                const int sr = ((hh >> 1) == r1) ? 1 : 0;
                const int sc = ((ww >> 1) == c1) ? 1 : 0;
                if (sr == 0 && sc == 0) wgt00 += mk;
                else if (sr == 0)       wgt01 += mk;
                else if (sc == 0)       wgt10 += mk;
                else                    wgt11 += mk;
            }
        }
    const float invden = 1.f / den;

    // ---- combine: out = num/den + relu(BN(conv2)) ----
    const float* ybase = y + (size_t)b * C * PIN;
    float* ob = out + (size_t)b * C * P + p;
#pragma unroll
    for (int v = 0; v < 8; ++v) {
        const int   o    = ct * 16 + v + half * 8;
        const float s    = g2[o] * rsqrtf(var2[o] + BN_EPS);
        const float bias = b2[o] * s + beta2[o] - m2[o] * s;
        const float c2v  = fmaxf(acc[v] + bias, 0.f);
        const float* yo  = ybase + o * PIN;
        const float num  = wgt00 * yo[r0c * WIN + c0c]
                         + wgt01 * yo[r0c * WIN + c1c]
                         + wgt10 * yo[r1c * WIN + c0c]
                         + wgt11 * yo[r1c * WIN + c1c];
        ob[o * P] = num * invden + c2v;
    }
}

// ---------------------------------------------------------------------------
extern "C" void kernel_launch(void* const* d_in, const int* in_sizes, int n_in,
                              void* d_out, int out_size, void* d_ws, size_t ws_size,
                              hipStream_t stream) {
    const float* x      = (const float*)d_in[0];
    const float* ref    = (const float*)d_in[1];
    const float* w1     = (const float*)d_in[2];
    const float* b1     = (const float*)d_in[3];
    const float* g1     = (const float*)d_in[4];
    const float* beta1  = (const float*)d_in[5];
    const float* m1     = (const float*)d_in[6];
    const float* var1   = (const float*)d_in[7];
    const float* w2     = (const float*)d_in[8];
    const float* b2     = (const float*)d_in[9];
    const float* g2     = (const float*)d_in[10];
    const float* beta2  = (const float*)d_in[11];
    const float* m2     = (const float*)d_in[12];
    const float* var2   = (const float*)d_in[13];
    float* out = (float*)d_out;

    // workspace layout: res (NB*P floats) then y (NB*C*PIN floats) ~ 8.9 MB
    float* res = (float*)d_ws;
    float* y   = res + (size_t)NB * P;

    // Kernel 0: channel mean of ref -> res. NB*P = 131072 threads.
    res_mean_kernel<<<(NB * P) / 256, 256, 0, stream>>>(ref, res);

    // Kernel 1: conv1+BN+ReLU. 8*4*256 = 8192 waves -> 1024 blocks of 256.
    conv1_wmma_kernel<<<1024, 256, 0, stream>>>(x, w1, b1, g1, beta1, m1, var1, y);

    // Kernel 2: conv2(WMMA)+mask+guided+add. 8*4*1024 = 32768 waves -> 4096 blocks.
    fused_out_kernel<<<4096, 256, 0, stream>>>(ref, w2, b2, g2, beta2, m2, var2,
                                               res, y, out);
}